// AgeSAGE_65163243815014
// MI455X (gfx1250) — compile-verified
//
#include <hip/hip_runtime.h>
#include <hip/hip_bf16.h>

typedef float v2f __attribute__((ext_vector_type(2)));
typedef float v8f __attribute__((ext_vector_type(8)));

#define HID 32
#define EPS 1e-5f

// ---------------------------------------------------------------------------
// Edge pass 1: degree + scatter-add of raw features (IN_DIM = 3)
// ---------------------------------------------------------------------------
__global__ __launch_bounds__(256) void k_scatter1(
    const int* __restrict__ src, const int* __restrict__ dst,
    const float* __restrict__ x, float* __restrict__ deg,
    float* __restrict__ agg1, int E)
{
    int e = blockIdx.x * blockDim.x + threadIdx.x;
    if (e >= E) return;
    int s = src[e], d = dst[e];
    atomicAdd(&deg[d], 1.0f);
    atomicAdd(&agg1[d * 3 + 0], x[s * 3 + 0]);
    atomicAdd(&agg1[d * 3 + 1], x[s * 3 + 1]);
    atomicAdd(&agg1[d * 3 + 2], x[s * 3 + 2]);
}

// ---------------------------------------------------------------------------
// Layer-1 SAGE linear (K=3, scalar FMAs) + BN1 statistics (LDS reduction)
// ---------------------------------------------------------------------------
__global__ __launch_bounds__(256) void k_layer1(
    const float* __restrict__ x, const float* __restrict__ agg1,
    const float* __restrict__ deg,
    const float* __restrict__ w1l, const float* __restrict__ b1l,
    const float* __restrict__ w1r,
    float* __restrict__ pre1, float* __restrict__ sum1, float* __restrict__ sq1,
    int N)
{
    __shared__ float sW[224];         // w1l[96] | w1r[96] | b1l[32]
    __shared__ float ssum[HID], ssq[HID];
    int tid = threadIdx.x;
    for (int i = tid; i < 96; i += blockDim.x) { sW[i] = w1l[i]; sW[96 + i] = w1r[i]; }
    if (tid < HID) { sW[192 + tid] = b1l[tid]; ssum[tid] = 0.f; ssq[tid] = 0.f; }
    __syncthreads();

    int n = blockIdx.x * blockDim.x + tid;
    if (n < N) {
        float invd = 1.0f / fmaxf(deg[n], 1.0f);
        float a0 = agg1[n * 3 + 0] * invd;
        float a1 = agg1[n * 3 + 1] * invd;
        float a2 = agg1[n * 3 + 2] * invd;
        float x0 = x[n * 3 + 0], x1 = x[n * 3 + 1], x2 = x[n * 3 + 2];
        #pragma unroll 8
        for (int j = 0; j < HID; ++j) {
            float v = sW[192 + j]
                    + a0 * sW[j] + a1 * sW[32 + j] + a2 * sW[64 + j]
                    + x0 * sW[96 + j] + x1 * sW[128 + j] + x2 * sW[160 + j];
            pre1[(size_t)n * HID + j] = v;
            atomicAdd(&ssum[j], v);
            atomicAdd(&ssq[j], v * v);
        }
    }
    __syncthreads();
    if (tid < HID) { atomicAdd(&sum1[tid], ssum[tid]); atomicAdd(&sq1[tid], ssq[tid]); }
}

// ---------------------------------------------------------------------------
// BatchNorm param derivation: scale/shift per column
// ---------------------------------------------------------------------------
__global__ void k_bnparams(const float* __restrict__ sum, const float* __restrict__ sq,
                           const float* __restrict__ gamma, const float* __restrict__ beta,
                           float* __restrict__ scale, float* __restrict__ shift, float n)
{
    int j = threadIdx.x;
    if (j >= HID) return;
    float mu  = sum[j] / n;
    float var = fmaxf(sq[j] / n - mu * mu, 0.0f);
    float sc  = gamma[j] * rsqrtf(var + EPS);
    scale[j] = sc;
    shift[j] = beta[j] - mu * sc;
}

// ---------------------------------------------------------------------------
// BN + ReLU elementwise apply (float4)
// ---------------------------------------------------------------------------
__global__ __launch_bounds__(256) void k_apply(
    const float* __restrict__ pre, const float* __restrict__ scale,
    const float* __restrict__ shift, float* __restrict__ h, int total4)
{
    int i = blockIdx.x * blockDim.x + threadIdx.x;
    if (i >= total4) return;
    int j = (i & 7) * 4;                       // column quad within 32-wide row
    float4 v = ((const float4*)pre)[i];
    v.x = fmaxf(v.x * scale[j + 0] + shift[j + 0], 0.f);
    v.y = fmaxf(v.y * scale[j + 1] + shift[j + 1], 0.f);
    v.z = fmaxf(v.z * scale[j + 2] + shift[j + 2], 0.f);
    v.w = fmaxf(v.w * scale[j + 3] + shift[j + 3], 0.f);
    ((float4*)h)[i] = v;
}

// ---------------------------------------------------------------------------
// Edge pass 2: scatter h1[src] into agg2[dst] (8 threads/edge, float4)
// ---------------------------------------------------------------------------
__global__ __launch_bounds__(256) void k_scatter2(
    const int* __restrict__ src, const int* __restrict__ dst,
    const float* __restrict__ h1, float* __restrict__ agg2, int E)
{
    int idx = blockIdx.x * blockDim.x + threadIdx.x;
    if (idx >= E * 8) return;
    int e = idx >> 3, q = idx & 7;
    int s = src[e], d = dst[e];
    float4 v = ((const float4*)(h1 + (size_t)s * HID))[q];
    float* p = agg2 + (size_t)d * HID + q * 4;
    atomicAdd(p + 0, v.x);
    atomicAdd(p + 1, v.y);
    atomicAdd(p + 2, v.z);
    atomicAdd(p + 3, v.w);
}

// ---------------------------------------------------------------------------
// Layer-2 SAGE linear via V_WMMA_F32_16X16X4_F32.
//   pre2 = [agg2/deg | h1] ([N x 64]) @ [w2l ; w2r] ([64 x 32]) + b2l
// One wave computes a 16x32 output tile (two 16x16 C frags, 32 WMMAs).
// BN2 statistics fused: each lane owns one output column in C layout.
// ---------------------------------------------------------------------------
__global__ __launch_bounds__(256) void k_sage2_wmma(
    const float* __restrict__ agg2, const float* __restrict__ h1,
    const float* __restrict__ deg,
    const float* __restrict__ w2l, const float* __restrict__ b2l,
    const float* __restrict__ w2r,
    float* __restrict__ pre2, float* __restrict__ sum2, float* __restrict__ sq2,
    int ntiles)
{
    const int lane   = threadIdx.x & 31;
    const int lane15 = lane & 15;
    const int khalf  = lane >> 4;            // 0: K={0,1}, 1: K={2,3} within a step
    const int wave   = blockIdx.x * (blockDim.x >> 5) + (threadIdx.x >> 5);
    const int nwaves = gridDim.x * (blockDim.x >> 5);

    // Preload all B fragments: 16 K-steps x 2 column tiles (64 VGPRs).
    // B 4x16 layout: lanes 0-15 hold K rows {k0, k0+1}, lanes 16-31 {k0+2, k0+3}.
    v2f bf[16][2];
    #pragma unroll
    for (int ks = 0; ks < 16; ++ks) {
        int kr = ks * 4 + 2 * khalf;                               // even, 0..62
        const float* Wp = (kr < 32) ? (w2l + kr * HID) : (w2r + (kr - 32) * HID);
        #pragma unroll
        for (int nt = 0; nt < 2; ++nt) {
            int col = nt * 16 + lane15;
            v2f b; b.x = Wp[col]; b.y = Wp[col + HID];
            bf[ks][nt] = b;
        }
    }
    const float bias0 = b2l[lane15];
    const float bias1 = b2l[16 + lane15];

    float s0 = 0.f, q0 = 0.f, s1 = 0.f, q1 = 0.f;

    for (int t = wave; t < ntiles; t += nwaves) {
        const int rowA = t * 16 + lane15;            // A row for this lane
        const float invd = 1.0f / fmaxf(deg[rowA], 1.0f);
        const float* ra = agg2 + (size_t)rowA * HID;
        const float* rh = h1   + (size_t)rowA * HID;

        v8f c0 = {};                                 // columns 0..15
        v8f c1 = {};                                 // columns 16..31
        #pragma unroll
        for (int ks = 0; ks < 16; ++ks) {
            v2f a;
            if (ks < 8) {                            // mean-aggregate half (K 0..31)
                int f = ks * 4 + 2 * khalf;
                a.x = ra[f] * invd; a.y = ra[f + 1] * invd;
            } else {                                 // root (h1) half (K 32..63)
                int f = (ks - 8) * 4 + 2 * khalf;
                a.x = rh[f]; a.y = rh[f + 1];
            }
            c0 = __builtin_amdgcn_wmma_f32_16x16x4_f32(false, a, false, bf[ks][0],
                                                       (short)0, c0, false, false);
            c1 = __builtin_amdgcn_wmma_f32_16x16x4_f32(false, a, false, bf[ks][1],
                                                       (short)0, c1, false, false);
        }

        // C layout: VGPR r -> rows {r, r+8}; lane15 selects column.
        float* prow = pre2 + (size_t)(t * 16 + 8 * khalf) * HID + lane15;
        #pragma unroll
        for (int r = 0; r < 8; ++r) {
            float v0 = c0[r] + bias0;
            float v1 = c1[r] + bias1;
            prow[r * HID]      = v0;
            prow[r * HID + 16] = v1;
            s0 += v0; q0 += v0 * v0;
            s1 += v1; q1 += v1 * v1;
        }
    }
    atomicAdd(&sum2[lane15],      s0);
    atomicAdd(&sq2[lane15],       q0);
    atomicAdd(&sum2[16 + lane15], s1);
    atomicAdd(&sq2[16 + lane15],  q1);
}

// Scalar tail for node rows not covered by full 16-row tiles (unused for N=100000).
__global__ void k_sage2_tail(
    const float* __restrict__ agg2, const float* __restrict__ h1,
    const float* __restrict__ deg,
    const float* __restrict__ w2l, const float* __restrict__ b2l,
    const float* __restrict__ w2r,
    float* __restrict__ pre2, float* __restrict__ sum2, float* __restrict__ sq2,
    int n0, int N)
{
    int n = n0 + blockIdx.x * blockDim.x + threadIdx.x;
    if (n >= N) return;
    float invd = 1.0f / fmaxf(deg[n], 1.0f);
    for (int j = 0; j < HID; ++j) {
        float v = b2l[j];
        for (int k = 0; k < HID; ++k)
            v += agg2[(size_t)n * HID + k] * invd * w2l[k * HID + j]
               + h1[(size_t)n * HID + k] * w2r[k * HID + j];
        pre2[(size_t)n * HID + j] = v;
        atomicAdd(&sum2[j], v);
        atomicAdd(&sq2[j], v * v);
    }
}

// ---------------------------------------------------------------------------
// Fused BN2 + ReLU + global mean-pool accumulation
// ---------------------------------------------------------------------------
__global__ __launch_bounds__(256) void k_pool(
    const float* __restrict__ pre2, const float* __restrict__ scale,
    const float* __restrict__ shift, const int* __restrict__ batch,
    float* __restrict__ gsum, float* __restrict__ gcnt, int N)
{
    int n = blockIdx.x * blockDim.x + threadIdx.x;
    if (n >= N) return;
    int g = batch[n];
    atomicAdd(&gcnt[g], 1.0f);
    const float4* row = (const float4*)(pre2 + (size_t)n * HID);
    float* gs = gsum + (size_t)g * HID;
    #pragma unroll
    for (int q = 0; q < 8; ++q) {
        float4 v = row[q];
        int j = q * 4;
        atomicAdd(&gs[j + 0], fmaxf(v.x * scale[j + 0] + shift[j + 0], 0.f));
        atomicAdd(&gs[j + 1], fmaxf(v.y * scale[j + 1] + shift[j + 1], 0.f));
        atomicAdd(&gs[j + 2], fmaxf(v.z * scale[j + 2] + shift[j + 2], 0.f));
        atomicAdd(&gs[j + 3], fmaxf(v.w * scale[j + 3] + shift[j + 3], 0.f));
    }
}

// ---------------------------------------------------------------------------
// Final readout: out[g] = mean_g(h2) @ wlin + blin
// ---------------------------------------------------------------------------
__global__ void k_final(const float* __restrict__ gsum, const float* __restrict__ gcnt,
                        const float* __restrict__ wlin, const float* __restrict__ blin,
                        float* __restrict__ out, int G)
{
    int g = blockIdx.x * blockDim.x + threadIdx.x;
    if (g >= G) return;
    float inv = 1.0f / fmaxf(gcnt[g], 1.0f);
    float acc = blin[0];
    for (int j = 0; j < HID; ++j)
        acc += gsum[(size_t)g * HID + j] * inv * wlin[j];
    out[g] = acc;
}

// ---------------------------------------------------------------------------
extern "C" void kernel_launch(void* const* d_in, const int* in_sizes, int n_in,
                              void* d_out, int out_size, void* d_ws, size_t ws_size,
                              hipStream_t stream)
{
    const float* x     = (const float*)d_in[0];
    const int*   ei    = (const int*)  d_in[1];
    const int*   batch = (const int*)  d_in[2];
    const float* w1l   = (const float*)d_in[3];
    const float* b1l   = (const float*)d_in[4];
    const float* w1r   = (const float*)d_in[5];
    const float* bn1_g = (const float*)d_in[6];
    const float* bn1_b = (const float*)d_in[7];
    const float* w2l   = (const float*)d_in[8];
    const float* b2l   = (const float*)d_in[9];
    const float* w2r   = (const float*)d_in[10];
    const float* bn2_g = (const float*)d_in[11];
    const float* bn2_b = (const float*)d_in[12];
    const float* wlin  = (const float*)d_in[13];
    const float* blin  = (const float*)d_in[14];
    float* out = (float*)d_out;

    const int N = in_sizes[0] / 3;
    const int E = in_sizes[1] / 2;
    const int G = out_size;
    const int* src = ei;
    const int* dst = ei + E;

    // Workspace layout (floats)
    float* ws     = (float*)d_ws;
    size_t Nz     = (size_t)N;
    float* deg    = ws;                 //  N
    float* agg1   = ws + Nz;            // 3N
    float* pre1   = ws + 4  * Nz;       // 32N
    float* h1     = ws + 36 * Nz;       // 32N
    float* agg2   = ws + 68 * Nz;       // 32N
    float* pre2   = ws + 100 * Nz;      // 32N
    float* stats  = ws + 132 * Nz;      // 256
    float* sum1   = stats;        float* sq1    = stats + 32;
    float* sum2   = stats + 64;   float* sq2    = stats + 96;
    float* scale1 = stats + 128;  float* shift1 = stats + 160;
    float* scale2 = stats + 192;  float* shift2 = stats + 224;
    float* gsum   = stats + 256;        // G*32
    float* gcnt   = gsum + (size_t)G * HID;  // G

    // Zero accumulators (async memsets are graph-capturable)
    hipMemsetAsync(deg,   0, 4  * Nz * sizeof(float), stream);   // deg + agg1
    hipMemsetAsync(agg2,  0, 32 * Nz * sizeof(float), stream);
    hipMemsetAsync(stats, 0, 128 * sizeof(float), stream);
    hipMemsetAsync(gsum,  0, ((size_t)G * HID + G) * sizeof(float), stream);

    const int B = 256;
    // Layer 1
    k_scatter1<<<(E + B - 1) / B, B, 0, stream>>>(src, dst, x, deg, agg1, E);
    k_layer1  <<<(N + B - 1) / B, B, 0, stream>>>(x, agg1, deg, w1l, b1l, w1r,
                                                  pre1, sum1, sq1, N);
    k_bnparams<<<1, 32, 0, stream>>>(sum1, sq1, bn1_g, bn1_b, scale1, shift1, (float)N);
    k_apply   <<<(N * 8 + B - 1) / B, B, 0, stream>>>(pre1, scale1, shift1, h1, N * 8);
    // Layer 2
    k_scatter2<<<(E * 8 + B - 1) / B, B, 0, stream>>>(src, dst, h1, agg2, E);
    const int ntiles = N / 16;
    k_sage2_wmma<<<256, B, 0, stream>>>(agg2, h1, deg, w2l, b2l, w2r,
                                        pre2, sum2, sq2, ntiles);
    if (N % 16) {
        int rem = N - ntiles * 16;
        k_sage2_tail<<<(rem + 63) / 64, 64, 0, stream>>>(agg2, h1, deg, w2l, b2l, w2r,
                                                         pre2, sum2, sq2, ntiles * 16, N);
    }
    k_bnparams<<<1, 32, 0, stream>>>(sum2, sq2, bn2_g, bn2_b, scale2, shift2, (float)N);
    // Pool + readout
    k_pool <<<(N + B - 1) / B, B, 0, stream>>>(pre2, scale2, shift2, batch, gsum, gcnt, N);
    k_final<<<(G + B - 1) / B, B, 0, stream>>>(gsum, gcnt, wlin, blin, out, G);
}